// TransE_47665547051802
// MI455X (gfx1250) — compile-verified
//
#include <hip/hip_runtime.h>
#include <hip/hip_bf16.h>
#include <stdint.h>

#define NUM_ENT   14541
#define NUM_REL2  474
#define EMBED_D   200
#define BATCH     256
#define GAMMA     9.0f

#define BT        32      // batch rows per workgroup
#define NT        64      // entity rows per tile
#define TILES_PER_WG 4
#define THREADS   256

#if defined(__gfx1250__) && __has_builtin(__builtin_amdgcn_tensor_load_to_lds) && __has_builtin(__builtin_amdgcn_s_wait_tensorcnt)
#define USE_TDM 1
#else
#define USE_TDM 0
#endif

#if USE_TDM
typedef unsigned int v4u_ __attribute__((ext_vector_type(4)));
typedef int          v8i_ __attribute__((ext_vector_type(8)));
typedef int          v4i_ __attribute__((ext_vector_type(4)));

// Issue one TDM 2D tile load: NT rows x EMBED_D f32 from gsrc (row stride
// EMBED_D) into LDS at ldsByteOff. rowsAvail clamps tensor_dim1 so the
// hardware zero-fills out-of-bounds rows on the tail tile.
__device__ __forceinline__ void tdm_load_rows(const float* gsrc,
                                              unsigned ldsByteOff,
                                              int rowsAvail)
{
    unsigned long long ga = (unsigned long long)(uintptr_t)gsrc;
    if (rowsAvail > 0xffff) rowsAvail = 0xffff;

    v4u_ g0;
    g0[0] = 1u;                                    // count=1, user descriptor, no gather
    g0[1] = ldsByteOff;                            // lds_addr (bytes)
    g0[2] = (unsigned)(ga & 0xffffffffull);        // global_addr[31:0]
    g0[3] = (unsigned)((ga >> 32) & 0x1ffffffull)  // global_addr[56:32]
          | (2u << 30);                            // type = 2

    v8i_ g1;
    g1[0] = 0x00020000;                                   // data_size=2 (4B); no mask/pad/iterate
    g1[1] = (int)((unsigned)EMBED_D << 16);               // tensor_dim0[15:0] = 200
    g1[2] = (int)(((unsigned)rowsAvail & 0xffffu) << 16); // tensor_dim1[15:0]
    g1[3] = (int)((unsigned)EMBED_D << 16);               // tile_dim0 = 200
    g1[4] = NT;                                           // tile_dim1 = NT, tile_dim2 = 0
    g1[5] = EMBED_D;                                      // tensor_dim0_stride = 200
    g1[6] = 0;
    g1[7] = 0;

    v4i_ z4 = {0, 0, 0, 0};                               // groups 2/3 unused (2D tile)
    v8i_ z8 = {0, 0, 0, 0, 0, 0, 0, 0};                   // toolchain's extra group
    __builtin_amdgcn_tensor_load_to_lds(g0, g1, z4, z4, z8, 0);
}
#endif

__device__ __forceinline__ float l1_4(float4 a, float4 b) {
    return __builtin_fabsf(a.x - b.x) + __builtin_fabsf(a.y - b.y)
         + __builtin_fabsf(a.z - b.z) + __builtin_fabsf(a.w - b.w);
}

__global__ __launch_bounds__(THREADS)
void transe_score_kernel(const float* __restrict__ ent,
                         const float* __restrict__ rel,
                         const int*   __restrict__ sub,
                         const int*   __restrict__ rels,
                         float*       __restrict__ out)
{
    __shared__ float s_obj[BT * EMBED_D];          // 25.6 KB
    __shared__ float s_ent[2][NT * EMBED_D];       // 2 x 51.2 KB double buffer

    const int tid    = threadIdx.x;
    const int b_base = blockIdx.y * BT;
    const int tile0  = blockIdx.x * TILES_PER_WG;

#if USE_TDM
    // Kick off the first DMA before building the obj tile so it overlaps.
    if (tid == 0) {
        int nb = tile0 * NT;
        if (nb < NUM_ENT)
            tdm_load_rows(ent + (size_t)nb * EMBED_D,
                          (unsigned)(size_t)(&s_ent[0][0]), NUM_ENT - nb);
    }
#endif

    // Build obj tile: obj[r][d] = ent[sub[b]][d] + rel[rels[b]][d]
    for (int idx = tid; idx < BT * EMBED_D; idx += THREADS) {
        int r = idx / EMBED_D;
        int d = idx - r * EMBED_D;
        int b = b_base + r;
        int si = sub[b];
        int ri = rels[b];
        s_obj[idx] = ent[(size_t)si * EMBED_D + d] + rel[(size_t)ri * EMBED_D + d];
    }

#if USE_TDM
    if (tid == 0) __builtin_amdgcn_s_wait_tensorcnt(0);
#endif
    __syncthreads();

    const int nx = tid & 15;        // entity lanes: rows nx + 16*j (stride-1 across lanes)
    const int by = tid >> 4;        // batch lanes:  rows by, by+16
    const float4* o0 = (const float4*)&s_obj[by * EMBED_D];
    const float4* o1 = (const float4*)&s_obj[(by + 16) * EMBED_D];
    float* outr0 = out + (size_t)(b_base + by)      * NUM_ENT;
    float* outr1 = out + (size_t)(b_base + by + 16) * NUM_ENT;

    for (int t = 0; t < TILES_PER_WG; ++t) {
        const int n_base = (tile0 + t) * NT;
        if (n_base >= NUM_ENT) break;            // uniform across block
        const int buf = t & 1;

#if USE_TDM
        // Prefetch next tile into the other buffer (overlaps with compute below).
        if (tid == 0) {
            int nb = (tile0 + t + 1) * NT;
            if ((t + 1) < TILES_PER_WG && nb < NUM_ENT)
                tdm_load_rows(ent + (size_t)nb * EMBED_D,
                              (unsigned)(size_t)(&s_ent[buf ^ 1][0]), NUM_ENT - nb);
        }
#else
        for (int idx = tid; idx < NT * EMBED_D; idx += THREADS) {
            int r = idx / EMBED_D;
            int d = idx - r * EMBED_D;
            int n = n_base + r;
            s_ent[buf][idx] = (n < NUM_ENT) ? ent[(size_t)n * EMBED_D + d] : 0.0f;
        }
        __syncthreads();
#endif

        const float4* e0 = (const float4*)&s_ent[buf][(nx     ) * EMBED_D];
        const float4* e1 = (const float4*)&s_ent[buf][(nx + 16) * EMBED_D];
        const float4* e2 = (const float4*)&s_ent[buf][(nx + 32) * EMBED_D];
        const float4* e3 = (const float4*)&s_ent[buf][(nx + 48) * EMBED_D];

        float a00 = 0.f, a01 = 0.f, a02 = 0.f, a03 = 0.f;
        float a10 = 0.f, a11 = 0.f, a12 = 0.f, a13 = 0.f;
#pragma unroll 5
        for (int k = 0; k < EMBED_D / 4; ++k) {
            float4 ov0 = o0[k];
            float4 ov1 = o1[k];
            float4 ev0 = e0[k];
            float4 ev1 = e1[k];
            float4 ev2 = e2[k];
            float4 ev3 = e3[k];
            a00 += l1_4(ov0, ev0);  a10 += l1_4(ov1, ev0);
            a01 += l1_4(ov0, ev1);  a11 += l1_4(ov1, ev1);
            a02 += l1_4(ov0, ev2);  a12 += l1_4(ov1, ev2);
            a03 += l1_4(ov0, ev3);  a13 += l1_4(ov1, ev3);
        }

        const int n0 = n_base + nx;
        if (n0 < NUM_ENT) {
            outr0[n0] = 1.0f / (1.0f + __expf(a00 - GAMMA));
            outr1[n0] = 1.0f / (1.0f + __expf(a10 - GAMMA));
        }
        if (n0 + 16 < NUM_ENT) {
            outr0[n0 + 16] = 1.0f / (1.0f + __expf(a01 - GAMMA));
            outr1[n0 + 16] = 1.0f / (1.0f + __expf(a11 - GAMMA));
        }
        if (n0 + 32 < NUM_ENT) {
            outr0[n0 + 32] = 1.0f / (1.0f + __expf(a02 - GAMMA));
            outr1[n0 + 32] = 1.0f / (1.0f + __expf(a12 - GAMMA));
        }
        if (n0 + 48 < NUM_ENT) {
            outr0[n0 + 48] = 1.0f / (1.0f + __expf(a03 - GAMMA));
            outr1[n0 + 48] = 1.0f / (1.0f + __expf(a13 - GAMMA));
        }

#if USE_TDM
        if (tid == 0) __builtin_amdgcn_s_wait_tensorcnt(0);
        __syncthreads();
#else
        __syncthreads();
#endif
    }
}

extern "C" void kernel_launch(void* const* d_in, const int* in_sizes, int n_in,
                              void* d_out, int out_size, void* d_ws, size_t ws_size,
                              hipStream_t stream) {
    const float* ent  = (const float*)d_in[0];   // [NUM_ENT, 200]
    const float* rel  = (const float*)d_in[1];   // [NUM_REL2, 200]
    const int*   sub  = (const int*)d_in[2];     // [256]
    const int*   rels = (const int*)d_in[3];     // [256]
    // d_in[4] = neg_ents: unused by the reference computation
    float* out = (float*)d_out;                  // [256, NUM_ENT]

    const int n_tiles = (NUM_ENT + NT - 1) / NT;                     // 228
    const int chunks  = (n_tiles + TILES_PER_WG - 1) / TILES_PER_WG; // 57
    dim3 grid(chunks, BATCH / BT, 1);
    dim3 block(THREADS, 1, 1);
    transe_score_kernel<<<grid, block, 0, stream>>>(ent, rel, sub, rels, out);
}